// VarunaGAT_83451214561470
// MI455X (gfx1250) — compile-verified
//
#include <hip/hip_runtime.h>
#include <hip/hip_bf16.h>
#include <math.h>

typedef __attribute__((ext_vector_type(2))) float v2f;
typedef __attribute__((ext_vector_type(8))) float v8f;

#define NNODES 50000
#define NEDGES 800000
#define ETOT   (NEDGES + NNODES)
#define NEG_SLOPE 0.2f

__device__ __forceinline__ float lrelu(float v) { return v > 0.f ? v : NEG_SLOPE * v; }
__device__ __forceinline__ float eluf(float v)  { return v > 0.f ? v : expm1f(v); }

// IEEE-ordered atomic float max (works for mixed signs; m initialized to -inf).
__device__ __forceinline__ void atomicMaxF(float* addr, float val) {
    if (val >= 0.f) atomicMax((int*)addr, __float_as_int(val));
    else            atomicMin((unsigned int*)addr, __float_as_uint(val));
}

// ---------------------------------------------------------------------------
// Build src/dst index arrays with appended self-loops (edge_index is int64).
// ---------------------------------------------------------------------------
__global__ void gat_build_edges(const long long* __restrict__ ei,
                                int* __restrict__ srcI, int* __restrict__ dstI,
                                int E, int n) {
    int i = blockIdx.x * blockDim.x + threadIdx.x;
    int etot = E + n;
    if (i >= etot) return;
    if (i < E) {
        srcI[i] = (int)ei[i];
        dstI[i] = (int)ei[E + i];
    } else {
        int v = i - E;
        srcI[i] = v;
        dstI[i] = v;
    }
}

// ---------------------------------------------------------------------------
// C[M,Cout] = A[M,K] @ W[K,Cout]  via V_WMMA_F32_16X16X4_F32.
// K, Cout are compile-time: strides fold, main K-loop is unguarded, the
// K%4 tail (layer 1 only, K=165) uses branchless clamped+masked loads so
// EXEC stays all-ones throughout (WMMA requirement).
// One wave computes one 16x16 C tile; 2 waves per block cover 32 columns.
// A frag (16x4):  lane L (ml=L&15, kp=(L>>4)*2) holds A[mT+ml][k0+kp+{0,1}]
// B frag (4x16):  lane L holds W[k0+kp+{0,1}][nT+ml]
// D/C (16x16):    reg r, lane L -> C[mT + r + (L>>4)*8][nT + (L&15)]
// ---------------------------------------------------------------------------
template <int K, int Cout>
__global__ void __launch_bounds__(64)
gat_gemm_wmma(const float* __restrict__ A, const float* __restrict__ W,
              float* __restrict__ C) {
    const int lane = threadIdx.x & 31;
    const int wave = threadIdx.x >> 5;
    const int mT = blockIdx.y * 16;
    const int nT = (blockIdx.x * 2 + wave) * 16;
    const int ml = lane & 15;
    const int kp = (lane >> 4) * 2;

    const float* __restrict__ arow = A + (size_t)(mT + ml) * K + kp;
    const float* __restrict__ wcol = W + (size_t)kp * Cout + nT + ml;

    v8f acc = {};
    constexpr int KM = K & ~3;      // unguarded main section

    int k0 = 0;
#pragma unroll 4
    for (; k0 < KM; k0 += 4) {
        __builtin_prefetch(arow + k0 + 64, 0, 1);             // global_prefetch_b8
        v2f a, b;
        a[0] = arow[k0];
        a[1] = arow[k0 + 1];
        b[0] = wcol[(size_t)k0 * Cout];
        b[1] = wcol[(size_t)(k0 + 1) * Cout];
        acc = __builtin_amdgcn_wmma_f32_16x16x4_f32(
            /*neg_a=*/false, a, /*neg_b=*/false, b,
            /*c_mod=*/(short)0, acc, /*reuse_a=*/false, /*reuse_b=*/false);
    }

    if constexpr (KM < K) {
        // tail: clamp index in-bounds, zero via 0/1 multiplier (v_cndmask, no branch)
        v2f a, b;
#pragma unroll
        for (int v = 0; v < 2; ++v) {
            int kk = KM + kp + v;
            int kc = (kk < K ? kk : K - 1) - kp;
            float msk = kk < K ? 1.f : 0.f;
            a[v] = arow[kc] * msk;
            b[v] = wcol[(size_t)kc * Cout] * msk;
        }
        acc = __builtin_amdgcn_wmma_f32_16x16x4_f32(
            false, a, false, b, (short)0, acc, false, false);
    }

#pragma unroll
    for (int r = 0; r < 8; ++r) {
        int mm = mT + r + (lane >> 4) * 8;
        C[(size_t)mm * Cout + nT + ml] = acc[r];
    }
}

// ---------------------------------------------------------------------------
// alpha_src / alpha_dst : per (node, head) dot of h[n,h,:] with a_{src,dst}[h,:]
// ---------------------------------------------------------------------------
__global__ void gat_alphas(const float* __restrict__ h,
                           const float* __restrict__ a_src,
                           const float* __restrict__ a_dst,
                           float* __restrict__ asrc, float* __restrict__ adst,
                           int n, int H, int C) {
    int i = blockIdx.x * blockDim.x + threadIdx.x;
    if (i >= n * H) return;
    int node = i / H, hd = i - node * H;
    const float* hp = h + (size_t)node * H * C + (size_t)hd * C;
    const float* as = a_src + hd * C;
    const float* ad = a_dst + hd * C;
    float s0 = 0.f, s1 = 0.f;
    for (int c = 0; c < C; ++c) {
        float v = hp[c];
        s0 += v * as[c];
        s1 += v * ad[c];
    }
    asrc[i] = s0;
    adst[i] = s1;
}

__global__ void gat_fill(float* __restrict__ p, float val, int count) {
    int i = blockIdx.x * blockDim.x + threadIdx.x;
    if (i < count) p[i] = val;
}

// ---------------------------------------------------------------------------
// Edge pass 1: segment max of leaky_relu(alpha_src[src]+alpha_dst[dst])
// ---------------------------------------------------------------------------
__global__ void gat_edge_max(const int* __restrict__ srcI, const int* __restrict__ dstI,
                             const float* __restrict__ asrc, const float* __restrict__ adst,
                             float* __restrict__ m, int etot, int H) {
    int i = blockIdx.x * blockDim.x + threadIdx.x;
    if (i >= etot * H) return;
    int e = i / H, hd = i - e * H;
    int s = srcI[e], d = dstI[e];
    float v = lrelu(asrc[s * H + hd] + adst[d * H + hd]);
    atomicMaxF(&m[d * H + hd], v);
}

// ---------------------------------------------------------------------------
// Edge pass 2: e = exp(v - m[dst]); store per-edge; segment sum into z
// ---------------------------------------------------------------------------
__global__ void gat_edge_exp(const int* __restrict__ srcI, const int* __restrict__ dstI,
                             const float* __restrict__ asrc, const float* __restrict__ adst,
                             const float* __restrict__ m, float* __restrict__ z,
                             float* __restrict__ eexp, int etot, int H) {
    int i = blockIdx.x * blockDim.x + threadIdx.x;
    if (i >= etot * H) return;
    int e = i / H, hd = i - e * H;
    int s = srcI[e], d = dstI[e];
    float v = lrelu(asrc[s * H + hd] + adst[d * H + hd]);
    float ee = __expf(v - m[d * H + hd]);
    eexp[i] = ee;
    atomicAdd(&z[d * H + hd], ee);
}

// ---------------------------------------------------------------------------
// Edge pass 3: out[dst] += h[src] * alpha. One block per edge, thread = channel.
// ---------------------------------------------------------------------------
__global__ void gat_edge_scatter(const int* __restrict__ srcI, const int* __restrict__ dstI,
                                 const float* __restrict__ h, const float* __restrict__ eexp,
                                 const float* __restrict__ z, float* __restrict__ out,
                                 int H, int C) {
    int e = blockIdx.x;
    int t = threadIdx.x;            // 0 .. H*C-1
    int hd = t / C;
    int s = srcI[e], d = dstI[e];
    float alpha = eexp[(size_t)e * H + hd] / (z[d * H + hd] + 1e-16f);
    atomicAdd(&out[(size_t)d * H * C + t], h[(size_t)s * H * C + t] * alpha);
}

// out[i] = elu(in[i] + b[i % cols])   (safe in place)
__global__ void gat_bias_elu(const float* __restrict__ in, const float* __restrict__ b,
                             float* __restrict__ out, int total, int cols) {
    int i = blockIdx.x * blockDim.x + threadIdx.x;
    if (i >= total) return;
    out[i] = eluf(in[i] + b[i % cols]);
}

// Fused: v = elu(agg + b3) ; out = v @ Wc + bc   (Wc is [64,2])
__global__ void gat_classifier(const float* __restrict__ agg, const float* __restrict__ b3,
                               const float* __restrict__ Wc, const float* __restrict__ bc,
                               float* __restrict__ out, int n) {
    int i = blockIdx.x * blockDim.x + threadIdx.x;
    if (i >= n) return;
    float o0 = bc[0], o1 = bc[1];
    const float* ap = agg + (size_t)i * 64;
    for (int c = 0; c < 64; ++c) {
        float v = eluf(ap[c] + b3[c]);
        o0 += v * Wc[c * 2 + 0];
        o1 += v * Wc[c * 2 + 1];
    }
    out[i * 2 + 0] = o0;
    out[i * 2 + 1] = o1;
}

// ---------------------------------------------------------------------------
extern "C" void kernel_launch(void* const* d_in, const int* in_sizes, int n_in,
                              void* d_out, int out_size, void* d_ws, size_t ws_size,
                              hipStream_t stream) {
    const float*     x   = (const float*)d_in[0];
    const long long* ei  = (const long long*)d_in[1];
    const float* W1  = (const float*)d_in[2];
    const float* a1s = (const float*)d_in[3];
    const float* a1d = (const float*)d_in[4];
    const float* b1  = (const float*)d_in[5];
    const float* W2  = (const float*)d_in[6];
    const float* a2s = (const float*)d_in[7];
    const float* a2d = (const float*)d_in[8];
    const float* b2  = (const float*)d_in[9];
    const float* W3  = (const float*)d_in[10];
    const float* a3s = (const float*)d_in[11];
    const float* a3d = (const float*)d_in[12];
    const float* b3  = (const float*)d_in[13];
    const float* Wc  = (const float*)d_in[14];
    const float* bc  = (const float*)d_in[15];
    float* out = (float*)d_out;

    const int N = NNODES, E = NEDGES, ET = ETOT;

    // workspace layout (floats)
    float* ws = (float*)d_ws;
    size_t off = 0;
    float* h    = ws + off; off += (size_t)N * 256;   // GEMM output (reused)
    float* agg  = ws + off; off += (size_t)N * 256;   // aggregation out == next input
    float* asrc = ws + off; off += (size_t)N * 4;
    float* adst = ws + off; off += (size_t)N * 4;
    float* mbuf = ws + off; off += (size_t)N * 4;
    float* zbuf = ws + off; off += (size_t)N * 4;
    float* eexp = ws + off; off += (size_t)ET * 4;
    int*   srcI = (int*)(ws + off); off += (size_t)ET;
    int*   dstI = (int*)(ws + off); off += (size_t)ET;
    (void)ws_size; (void)n_in; (void)in_sizes; (void)out_size;

    const int TB = 256;
    auto nb = [](long long total, int tb) { return (int)((total + tb - 1) / tb); };

    // edge lists with self loops
    gat_build_edges<<<nb(ET, TB), TB, 0, stream>>>(ei, srcI, dstI, E, N);

    const float NEG_INF = -__builtin_huge_valf();

    // ---- Layer 1: in=x [N,165] -> h [N,256], H=4, C=64 --------------------
    gat_gemm_wmma<165, 256><<<dim3(256 / 32, N / 16), 64, 0, stream>>>(x, W1, h);
    gat_alphas<<<nb((long long)N * 4, TB), TB, 0, stream>>>(h, a1s, a1d, asrc, adst, N, 4, 64);
    gat_fill<<<nb((long long)N * 4, TB), TB, 0, stream>>>(mbuf, NEG_INF, N * 4);
    gat_fill<<<nb((long long)N * 4, TB), TB, 0, stream>>>(zbuf, 0.f, N * 4);
    gat_fill<<<nb((long long)N * 256, TB), TB, 0, stream>>>(agg, 0.f, N * 256);
    gat_edge_max<<<nb((long long)ET * 4, TB), TB, 0, stream>>>(srcI, dstI, asrc, adst, mbuf, ET, 4);
    gat_edge_exp<<<nb((long long)ET * 4, TB), TB, 0, stream>>>(srcI, dstI, asrc, adst, mbuf, zbuf, eexp, ET, 4);
    gat_edge_scatter<<<ET, 256, 0, stream>>>(srcI, dstI, h, eexp, zbuf, agg, 4, 64);
    gat_bias_elu<<<nb((long long)N * 256, TB), TB, 0, stream>>>(agg, b1, agg, N * 256, 256);

    // ---- Layer 2: in=agg [N,256] -> h [N,256], H=4, C=64 ------------------
    gat_gemm_wmma<256, 256><<<dim3(256 / 32, N / 16), 64, 0, stream>>>(agg, W2, h);
    gat_alphas<<<nb((long long)N * 4, TB), TB, 0, stream>>>(h, a2s, a2d, asrc, adst, N, 4, 64);
    gat_fill<<<nb((long long)N * 4, TB), TB, 0, stream>>>(mbuf, NEG_INF, N * 4);
    gat_fill<<<nb((long long)N * 4, TB), TB, 0, stream>>>(zbuf, 0.f, N * 4);
    gat_fill<<<nb((long long)N * 256, TB), TB, 0, stream>>>(agg, 0.f, N * 256);
    gat_edge_max<<<nb((long long)ET * 4, TB), TB, 0, stream>>>(srcI, dstI, asrc, adst, mbuf, ET, 4);
    gat_edge_exp<<<nb((long long)ET * 4, TB), TB, 0, stream>>>(srcI, dstI, asrc, adst, mbuf, zbuf, eexp, ET, 4);
    gat_edge_scatter<<<ET, 256, 0, stream>>>(srcI, dstI, h, eexp, zbuf, agg, 4, 64);
    gat_bias_elu<<<nb((long long)N * 256, TB), TB, 0, stream>>>(agg, b2, agg, N * 256, 256);

    // ---- Layer 3: in=agg [N,256] -> h [N,64], H=1, C=64 -------------------
    gat_gemm_wmma<256, 64><<<dim3(64 / 32, N / 16), 64, 0, stream>>>(agg, W3, h);
    gat_alphas<<<nb((long long)N, TB), TB, 0, stream>>>(h, a3s, a3d, asrc, adst, N, 1, 64);
    gat_fill<<<nb((long long)N, TB), TB, 0, stream>>>(mbuf, NEG_INF, N);
    gat_fill<<<nb((long long)N, TB), TB, 0, stream>>>(zbuf, 0.f, N);
    gat_fill<<<nb((long long)N * 64, TB), TB, 0, stream>>>(agg, 0.f, N * 64);
    gat_edge_max<<<nb((long long)ET, TB), TB, 0, stream>>>(srcI, dstI, asrc, adst, mbuf, ET, 1);
    gat_edge_exp<<<nb((long long)ET, TB), TB, 0, stream>>>(srcI, dstI, asrc, adst, mbuf, zbuf, eexp, ET, 1);
    gat_edge_scatter<<<ET, 64, 0, stream>>>(srcI, dstI, h, eexp, zbuf, agg, 1, 64);

    // Classifier: elu(agg + b3) @ Wc + bc -> [N,2]
    gat_classifier<<<nb((long long)N, TB), TB, 0, stream>>>(agg, b3, Wc, bc, out, N);
}